// LBLHighwayBiLm_62689342652479
// MI455X (gfx1250) — compile-verified
//
#include <hip/hip_runtime.h>
#include <hip/hip_bf16.h>

#define B_  32
#define S_  1024
#define D_  256
#define W_  4
#define NL_ 2
#define NH_ 2

typedef __attribute__((ext_vector_type(2))) float v2f;
typedef __attribute__((ext_vector_type(8))) float v8f;

// ---------------------------------------------------------------------------
// FIR tap filter along S with learned front/back padding.
// y[b,s,d] = sum_j w[j] * pad_or_x(t = s + off + j)
//   t <  W        -> front pad (fwd_pad[layer])
//   t >= W+S      -> back  pad (bwd_pad[layer])
//   else          -> x[b, t-W, d]   (row stride xrow: D for layer0, 2D after)
// ---------------------------------------------------------------------------
__global__ __launch_bounds__(256)
void fir_kernel(const float* __restrict__ x, int xrow,
                const float* __restrict__ padf, const float* __restrict__ padb,
                const float* __restrict__ w, int off,
                float* __restrict__ y) {
  int gid = blockIdx.x * blockDim.x + threadIdx.x;   // = ((b*S + s)*D + d)
  int d = gid & (D_ - 1);
  int s = (gid >> 8) & (S_ - 1);
  int b = gid >> 18;
  float acc = 0.f;
#pragma unroll
  for (int j = 0; j <= W_; ++j) {
    int t = s + off + j;
    float v;
    if (t < W_)             v = padf[t * D_ + d];
    else if (t >= W_ + S_)  v = padb[(t - W_ - S_) * D_ + d];
    else                    v = x[((size_t)b * S_ + (t - W_)) * (size_t)xrow + d];
    acc = fmaf(w[j], v, acc);
  }
  y[(size_t)gid] = acc;
}

// ---------------------------------------------------------------------------
// Fused Highway layer: proj = x @ W^T + bias, split (nonlin, gate),
// out = sigmoid(gate)*x + (1-sigmoid(gate))*relu(nonlin).
// x: [M, 256] contiguous. W: [512, 256] row-major (rows 0..255 = nonlin,
// 256..511 = gate). out row stride `orow` (256 for scratch, 512 into d_out).
//
// Per wave: 32(M) x 32(N) output tile -> 2 A fragments, 4 B fragments,
// 8 f32 accumulators; K=256 via 64 K-groups of V_WMMA_F32_16X16X4_F32
// (8 wmma per 6 b64 loads).
// ---------------------------------------------------------------------------
__global__ __launch_bounds__(256)
void highway_kernel(const float* __restrict__ x,
                    const float* __restrict__ Wt,
                    const float* __restrict__ bias,
                    float* __restrict__ out, int orow) {
  const int lane = threadIdx.x & 31;
  const int wave = threadIdx.x >> 5;
  const int half = lane >> 4;      // K-split: low lanes K+{0,1}, high lanes K+{2,3}
  const int l    = lane & 15;

  const int m0 = (blockIdx.y * 8 + wave) * 32;   // 32 M rows per wave
  const int n0 = blockIdx.x * 32;                // 32 N cols per wave
  const int nA = n0 + l;                         // N-subtile 0 column
  const int nB = n0 + 16 + l;                    // N-subtile 1 column

  const float* arow0 = x  + (size_t)(m0 + l) * D_       + 2 * half;  // A frag, M 0..15
  const float* arow1 = x  + (size_t)(m0 + 16 + l) * D_  + 2 * half;  // A frag, M 16..31
  const float* wn0   = Wt + (size_t)nA * D_             + 2 * half;  // B nonlin
  const float* wn1   = Wt + (size_t)nB * D_             + 2 * half;
  const float* wg0   = Wt + (size_t)(D_ + nA) * D_      + 2 * half;  // B gate
  const float* wg1   = Wt + (size_t)(D_ + nB) * D_      + 2 * half;

  v8f aN0 = {}; v8f aN1 = {}; v8f aG0 = {}; v8f aG1 = {};   // M-subtile 0
  v8f bN0 = {}; v8f bN1 = {}; v8f bG0 = {}; v8f bG1 = {};   // M-subtile 1

#pragma unroll 2
  for (int g = 0; g < D_ / 4; ++g) {
    v2f a0 = *(const v2f*)(arow0 + 4 * g);
    v2f a1 = *(const v2f*)(arow1 + 4 * g);
    v2f n0v = *(const v2f*)(wn0 + 4 * g);
    v2f n1v = *(const v2f*)(wn1 + 4 * g);
    v2f g0v = *(const v2f*)(wg0 + 4 * g);
    v2f g1v = *(const v2f*)(wg1 + 4 * g);
    aN0 = __builtin_amdgcn_wmma_f32_16x16x4_f32(false, a0, false, n0v, (short)0, aN0, false, false);
    aN1 = __builtin_amdgcn_wmma_f32_16x16x4_f32(false, a0, false, n1v, (short)0, aN1, false, false);
    aG0 = __builtin_amdgcn_wmma_f32_16x16x4_f32(false, a0, false, g0v, (short)0, aG0, false, false);
    aG1 = __builtin_amdgcn_wmma_f32_16x16x4_f32(false, a0, false, g1v, (short)0, aG1, false, false);
    bN0 = __builtin_amdgcn_wmma_f32_16x16x4_f32(false, a1, false, n0v, (short)0, bN0, false, false);
    bN1 = __builtin_amdgcn_wmma_f32_16x16x4_f32(false, a1, false, n1v, (short)0, bN1, false, false);
    bG0 = __builtin_amdgcn_wmma_f32_16x16x4_f32(false, a1, false, g0v, (short)0, bG0, false, false);
    bG1 = __builtin_amdgcn_wmma_f32_16x16x4_f32(false, a1, false, g1v, (short)0, bG1, false, false);
  }

  const float bnA = bias[nA], bnB = bias[nB];
  const float bgA = bias[D_ + nA], bgB = bias[D_ + nB];

  // C/D layout: VGPR r -> row mbase+r (lanes 0-15) / mbase+8+r (lanes 16-31),
  // col = lane&15. Fused bias + sigmoid gate + relu + residual epilogue.
#pragma unroll
  for (int r = 0; r < 8; ++r) {
    {
      const int m = m0 + r + 8 * half;           // M-subtile 0
      const float* xr = x + (size_t)m * D_;
      float* op = out + (size_t)m * orow;
      float g0 = 1.f / (1.f + __expf(-(aG0[r] + bgA)));
      op[nA] = g0 * xr[nA] + (1.f - g0) * fmaxf(aN0[r] + bnA, 0.f);
      float g1 = 1.f / (1.f + __expf(-(aG1[r] + bgB)));
      op[nB] = g1 * xr[nB] + (1.f - g1) * fmaxf(aN1[r] + bnB, 0.f);
    }
    {
      const int m = m0 + 16 + r + 8 * half;      // M-subtile 1
      const float* xr = x + (size_t)m * D_;
      float* op = out + (size_t)m * orow;
      float g0 = 1.f / (1.f + __expf(-(bG0[r] + bgA)));
      op[nA] = g0 * xr[nA] + (1.f - g0) * fmaxf(bN0[r] + bnA, 0.f);
      float g1 = 1.f / (1.f + __expf(-(bG1[r] + bgB)));
      op[nB] = g1 * xr[nB] + (1.f - g1) * fmaxf(bN1[r] + bnB, 0.f);
    }
  }
}

// ---------------------------------------------------------------------------
extern "C" void kernel_launch(void* const* d_in, const int* in_sizes, int n_in,
                              void* d_out, int out_size, void* d_ws, size_t ws_size,
                              hipStream_t stream) {
  (void)in_sizes; (void)n_in; (void)out_size; (void)ws_size;

  const float* inputs   = (const float*)d_in[0];  // [B,S,D]
  const float* fwd_pad  = (const float*)d_in[1];  // [L,W,D]
  const float* bwd_pad  = (const float*)d_in[2];  // [L,W,D]
  const float* fwd_w    = (const float*)d_in[3];  // [L,W+1]
  const float* bwd_w    = (const float*)d_in[4];  // [L,W+1]
  const float* fwd_hw_W = (const float*)d_in[5];  // [L,H,2D,D]
  const float* fwd_hw_b = (const float*)d_in[6];  // [L,H,2D]
  const float* bwd_hw_W = (const float*)d_in[7];
  const float* bwd_hw_b = (const float*)d_in[8];
  float* out = (float*)d_out;                     // [L,B,S,2D]

  const size_t M = (size_t)B_ * S_;
  float* yA = (float*)d_ws;                       // [M, D] FIR output
  float* yB = yA + M * D_;                        // [M, D] highway-1 output

  dim3 hwBlock(256);
  dim3 hwGrid(D_ / 32, (unsigned)(M / (32 * 8)));   // 32 M-rows per wave, 8 waves/block
  const int firBlocks = (int)(M * D_ / 256);

  for (int i = 0; i < NL_; ++i) {
    for (int dir = 0; dir < 2; ++dir) {
      const float* xin; int xrow;
      if (i == 0) { xin = inputs; xrow = D_; }
      else        { xin = out + (size_t)(i - 1) * M * 2 * D_ + dir * D_; xrow = 2 * D_; }

      const float* taps = (dir ? bwd_w : fwd_w) + i * (W_ + 1);
      fir_kernel<<<firBlocks, 256, 0, stream>>>(
          xin, xrow,
          fwd_pad + (size_t)i * W_ * D_, bwd_pad + (size_t)i * W_ * D_,
          taps, dir ? W_ : 0, yA);

      const float* HW = (dir ? bwd_hw_W : fwd_hw_W) + (size_t)i * NH_ * 2 * D_ * D_;
      const float* Hb = (dir ? bwd_hw_b : fwd_hw_b) + (size_t)i * NH_ * 2 * D_;

      // highway layer 0: yA -> yB
      highway_kernel<<<hwGrid, hwBlock, 0, stream>>>(yA, HW, Hb, yB, D_);
      // highway layer 1: yB -> d_out[i, :, :, dir*D : dir*D+D]
      highway_kernel<<<hwGrid, hwBlock, 0, stream>>>(
          yB, HW + (size_t)2 * D_ * D_, Hb + 2 * D_,
          out + (size_t)i * M * 2 * D_ + dir * D_, 2 * D_);
    }
  }
}